// LinearChainCRF_31172872634756
// MI455X (gfx1250) — compile-verified
//
#include <hip/hip_runtime.h>

// Linear-chain CRF NLL for B=128, S=8192, L=32 on gfx1250 (MI455X).
// Denominator = chunked log-semiring scan; per-timestep 32x32x32 GEMM via
// 32x V_WMMA_F32_16X16X4_F32 per wave. 2048 independent chunk waves.
// Raw v_exp_f32 / v_log_f32 (no OCML denormal fixup) for the lse exp/log.

#define B_  128
#define S_  8192
#define L_  32
#define NC_ 16           // chunks per batch
#define TC_ 512          // timesteps per chunk (transitions)

typedef __attribute__((ext_vector_type(2))) float v2f;
typedef __attribute__((ext_vector_type(8))) float v8f;

#define LOG2E 1.44269504088896340736f
#define LN2   0.69314718055994530942f
#define NEG_BIG (-1e30f)

// ds_swizzle group-of-32: offset = xor<<10 | or<<5 | and ; and=0x1f
#define SWZ(x, off) __int_as_float(__builtin_amdgcn_ds_swizzle(__float_as_int(x), (off)))
#define SWAPX16_OFF 0x401F

// Raw hardware transcendentals: v_exp_f32 (2^x), v_log_f32 (log2 x).
__device__ __forceinline__ float hexp2(float x) { return __builtin_amdgcn_exp2f(x); }
__device__ __forceinline__ float hlog2(float x) { return __builtin_amdgcn_logf(x); }
__device__ __forceinline__ float fexp(float x) { return hexp2(x * LOG2E); }
__device__ __forceinline__ float flog(float x) { return hlog2(x) * LN2; }

// ---------------------------------------------------------------------------
// Phase 1: per (batch, chunk) wave computes the 32x32 log-semiring transfer
// matrix G[j][i] (j = state after chunk, i = state before chunk), including
// emissions, using f32 WMMA for the per-step 32x32x32 GEMM.
// ---------------------------------------------------------------------------
__global__ __launch_bounds__(32)
void crf_chunk_kernel(const float* __restrict__ logits,
                      const unsigned char* __restrict__ mask,
                      const float* __restrict__ trans,
                      float* __restrict__ wsMat) {
  const int lane = threadIdx.x;
  const int ln16 = lane & 15;
  const bool hi  = lane >= 16;
  const int wg = blockIdx.x;
  const int b  = wg >> 4;          // / NC_
  const int c  = wg & (NC_ - 1);

  // Constant A-operand fragments: E^T, EA[mj][f] elem v = exp(T[k][j]),
  // A 16x4 layout: M = lane%16 (j = mj*16+ln16), K = 4f + v + 2h.
  v2f EA[2][8];
#pragma unroll
  for (int mj = 0; mj < 2; ++mj) {
#pragma unroll
    for (int f = 0; f < 8; ++f) {
      const int j  = mj * 16 + ln16;
      const int k0 = 4 * f + (hi ? 2 : 0);
      EA[mj][f].x = fexp(trans[k0 * L_ + j]);
      EA[mj][f].y = fexp(trans[(k0 + 1) * L_ + j]);
    }
  }

  // G in C/D layout: G[rj][ci][r] holds (j = rj*16 + r + 8h, i = ci*16 + ln16).
  // Init: log-semiring identity.
  float G[2][2][8];
#pragma unroll
  for (int rj = 0; rj < 2; ++rj)
#pragma unroll
    for (int ci = 0; ci < 2; ++ci)
#pragma unroll
      for (int r = 0; r < 8; ++r) {
        const int j = rj * 16 + r + (hi ? 8 : 0);
        const int i = ci * 16 + ln16;
        G[rj][ci][r] = (i == j) ? 0.0f : NEG_BIG;
      }

  const float* logitsB        = logits + (size_t)b * S_ * L_;
  const unsigned char* maskB  = mask + (size_t)b * S_;
  const int t0 = c * TC_ + 1;
  const int t1 = (c == NC_ - 1) ? (S_ - 1) : (c + 1) * TC_;
  const int hoff = hi ? 32 : 0;    // bpermute byte offset for the +8h row

  for (int t = t0; t <= t1; ++t) {
    const float lv = logitsB[t * L_ + lane];       // emission row, lane = state
    if (t + 8 <= t1)
      __builtin_prefetch(&logitsB[(t + 8) * L_ + lane], 0, 0);
    if (!maskB[t]) continue;                       // wave-uniform: skip step

    // ---- global max of G (exp-trick shift) ----
    float m = NEG_BIG;
#pragma unroll
    for (int rj = 0; rj < 2; ++rj)
#pragma unroll
      for (int ci = 0; ci < 2; ++ci)
#pragma unroll
        for (int r = 0; r < 8; ++r) m = fmaxf(m, G[rj][ci][r]);
    m = fmaxf(m, SWZ(m, 0x041F));   // xor 1
    m = fmaxf(m, SWZ(m, 0x081F));   // xor 2
    m = fmaxf(m, SWZ(m, 0x101F));   // xor 4
    m = fmaxf(m, SWZ(m, 0x201F));   // xor 8
    m = fmaxf(m, SWZ(m, 0x401F));   // xor 16 (cross half)

    const float negms = -m * LOG2E;  // exp2 shift, folded into one FMA/elem

    // ---- build B operand: Bf[f][ci] elem v = exp(G[k][i] - m),
    //      B 4x16 layout: K = 4f + v + 2h, N = lane%16.
    //      Source element lives in G[.][.][k&7] at lane-half k>>3:
    //      one half is same-lane, the other arrives via SWAPX16.
    v2f Bf[8][2];
#pragma unroll
    for (int f = 0; f < 8; ++f) {
      const int rj = f >> 2;
      const int fl = f & 3;
#pragma unroll
      for (int ci = 0; ci < 2; ++ci) {
#pragma unroll
        for (int v = 0; v < 2; ++v) {
          const int k0 = 4 * fl + v;       // dest half h=0 local K
          const int k1 = k0 + 2;           // dest half h=1 local K
          float c0, c1;
          if (fl < 2) {                    // sources live in half 0
            c0 = G[rj][ci][k0 & 7];                    // same-lane
            c1 = SWZ(G[rj][ci][k1 & 7], SWAPX16_OFF);  // cross
          } else {                         // sources live in half 1
            c0 = SWZ(G[rj][ci][k0 & 7], SWAPX16_OFF);  // cross
            c1 = G[rj][ci][k1 & 7];                    // same-lane
          }
          const float mg = hi ? c1 : c0;
          const float p  = hexp2(__builtin_fmaf(mg, LOG2E, negms));
          if (v == 0) Bf[f][ci].x = p; else Bf[f][ci].y = p;
        }
      }
    }

    // ---- Q = E^T * P : 2 M-tiles x 2 N-tiles x 8 K-frags = 32 WMMAs ----
    v8f D[2][2];
#pragma unroll
    for (int mj = 0; mj < 2; ++mj)
#pragma unroll
      for (int ci = 0; ci < 2; ++ci)
#pragma unroll
        for (int e = 0; e < 8; ++e) D[mj][ci][e] = 0.0f;

#pragma unroll
    for (int f = 0; f < 8; ++f) {
#pragma unroll
      for (int mj = 0; mj < 2; ++mj) {
#pragma unroll
        for (int ci = 0; ci < 2; ++ci) {
          D[mj][ci] = __builtin_amdgcn_wmma_f32_16x16x4_f32(
              false, EA[mj][f], false, Bf[f][ci], (short)0, D[mj][ci],
              false, false);
        }
      }
    }

    // ---- G' = log(Q) + m + logit_t[j] (row-constant via ds_bpermute) ----
#pragma unroll
    for (int rj = 0; rj < 2; ++rj) {
#pragma unroll
      for (int r = 0; r < 8; ++r) {
        const int idx = (rj * 16 + r) * 4 + hoff;   // byte index of state j
        const float lrow =
            __int_as_float(__builtin_amdgcn_ds_bpermute(idx, __float_as_int(lv)));
        const float base = m + lrow;
#pragma unroll
        for (int ci = 0; ci < 2; ++ci)
          G[rj][ci][r] = __builtin_fmaf(hlog2(D[rj][ci][r]), LN2, base);
      }
    }
  }

  // ---- store transfer matrix row-major [j][i] (coalesced per half) ----
  float* out = wsMat + ((size_t)(b * NC_ + c) << 10);
#pragma unroll
  for (int rj = 0; rj < 2; ++rj)
#pragma unroll
    for (int ci = 0; ci < 2; ++ci)
#pragma unroll
      for (int r = 0; r < 8; ++r) {
        const int j = rj * 16 + r + (hi ? 8 : 0);
        const int i = ci * 16 + ln16;
        out[j * L_ + i] = G[rj][ci][r];
      }
}

// ---------------------------------------------------------------------------
// Phase 2: one wave per batch; fold the NC_ chunk matrices sequentially:
// alpha'_j = lse_i(alpha_i + Gc[j][i]); then den = lse_j(alpha_j + end_j).
// ---------------------------------------------------------------------------
__global__ __launch_bounds__(32)
void crf_combine_kernel(const float* __restrict__ logits,
                        const float* __restrict__ startT,
                        const float* __restrict__ endT,
                        const float* __restrict__ wsMat,
                        float* __restrict__ wsDen) {
  const int b = blockIdx.x;
  const int lane = threadIdx.x;           // lane = state j
  float alpha = startT[lane] + logits[(size_t)b * S_ * L_ + lane];

  for (int c = 0; c < NC_; ++c) {
    const float* M = wsMat + ((size_t)(b * NC_ + c) << 10) + lane * L_;
    float vals[L_];
    float mx = NEG_BIG;
#pragma unroll
    for (int i = 0; i < L_; ++i) {
      vals[i] = __shfl(alpha, i, 32) + M[i];
      mx = fmaxf(mx, vals[i]);
    }
    const float negms = -mx * LOG2E;
    float s = 0.0f;
#pragma unroll
    for (int i = 0; i < L_; ++i)
      s += hexp2(__builtin_fmaf(vals[i], LOG2E, negms));
    alpha = __builtin_fmaf(hlog2(s), LN2, mx);
  }

  float v = alpha + endT[lane];
  float mx = v;
#pragma unroll
  for (int d = 1; d < 32; d <<= 1) mx = fmaxf(mx, __shfl_xor(mx, d, 32));
  float s = hexp2((v - mx) * LOG2E);
#pragma unroll
  for (int d = 1; d < 32; d <<= 1) s += __shfl_xor(s, d, 32);
  if (lane == 0) wsDen[b] = __builtin_fmaf(hlog2(s), LN2, mx);
}

// ---------------------------------------------------------------------------
// Numerator: one block per batch, grid-stride gather + reduce.
// ---------------------------------------------------------------------------
__global__ __launch_bounds__(256)
void crf_numer_kernel(const float* __restrict__ logits,
                      const int* __restrict__ labels,
                      const unsigned char* __restrict__ mask,
                      const float* __restrict__ trans,
                      const float* __restrict__ startT,
                      const float* __restrict__ endT,
                      float* __restrict__ wsNum) {
  __shared__ float sacc[256];
  __shared__ int   scnt[256];
  const int b = blockIdx.x, tid = threadIdx.x;
  const float* lg = logits + (size_t)b * S_ * L_;
  const int* lb = labels + (size_t)b * S_;
  const unsigned char* mk = mask + (size_t)b * S_;

  float acc = 0.0f;
  int cnt = 0;
  for (int t = tid; t < S_; t += 256) {
    const int mt = mk[t] ? 1 : 0;
    cnt += mt;
    if (t < S_ - 1) {
      const int lt = lb[t], lt1 = lb[t + 1];
      const int mt1 = mk[t + 1] ? 1 : 0;
      acc += lg[t * L_ + lt] * (float)mt + trans[lt * L_ + lt1] * (float)mt1;
    }
  }
  sacc[tid] = acc;
  scnt[tid] = cnt;
  __syncthreads();
  for (int s = 128; s > 0; s >>= 1) {
    if (tid < s) { sacc[tid] += sacc[tid + s]; scnt[tid] += scnt[tid + s]; }
    __syncthreads();
  }
  if (tid == 0) {
    int last = scnt[0] - 1;
    if (last < 0) last = 0;
    const int ll = lb[last];
    const float sc = startT[lb[0]] + sacc[0] +
                     lg[(S_ - 1) * L_ + ll] * (mk[S_ - 1] ? 1.0f : 0.0f) +
                     endT[ll];
    wsNum[b] = sc;
  }
}

// ---------------------------------------------------------------------------
// Final: loss = -mean(num - den)
// ---------------------------------------------------------------------------
__global__ __launch_bounds__(128)
void crf_loss_kernel(const float* __restrict__ wsNum,
                     const float* __restrict__ wsDen,
                     float* __restrict__ out) {
  __shared__ float sm[128];
  const int tid = threadIdx.x;
  sm[tid] = wsNum[tid] - wsDen[tid];
  __syncthreads();
  for (int s = 64; s > 0; s >>= 1) {
    if (tid < s) sm[tid] += sm[tid + s];
    __syncthreads();
  }
  if (tid == 0) out[0] = -sm[0] / (float)B_;
}

extern "C" void kernel_launch(void* const* d_in, const int* in_sizes, int n_in,
                              void* d_out, int out_size, void* d_ws, size_t ws_size,
                              hipStream_t stream) {
  (void)in_sizes; (void)n_in; (void)out_size; (void)ws_size;
  const float* logits         = (const float*)d_in[0];
  const int* labels           = (const int*)d_in[1];
  const unsigned char* maskp  = (const unsigned char*)d_in[2];  // jnp bool = 1 byte
  const float* trans          = (const float*)d_in[3];
  const float* startT         = (const float*)d_in[4];
  const float* endT           = (const float*)d_in[5];

  float* wsf   = (float*)d_ws;
  float* wsNum = wsf;                 // [B_]
  float* wsDen = wsf + B_;            // [B_]
  float* wsMat = wsf + 2 * B_;        // [B_ * NC_ * 1024]
  float* out   = (float*)d_out;

  crf_numer_kernel<<<B_, 256, 0, stream>>>(logits, labels, maskp, trans,
                                           startT, endT, wsNum);
  crf_chunk_kernel<<<B_ * NC_, 32, 0, stream>>>(logits, maskp, trans, wsMat);
  crf_combine_kernel<<<B_, 32, 0, stream>>>(logits, startT, endT, wsMat, wsDen);
  crf_loss_kernel<<<1, 128, 0, stream>>>(wsNum, wsDen, out);
}